// TAN_22007412425058
// MI455X (gfx1250) — compile-verified
//
#include <hip/hip_runtime.h>
#include <hip/hip_bf16.h>
#include <stdint.h>

// Problem constants (fixed by the reference).
#define B_    64
#define S_    16
#define N_    256
#define T_    256
#define D_    500
#define NF_   16
#define TOPK_ 3

typedef unsigned int u32;
typedef u32 v4u_ __attribute__((ext_vector_type(4)));
typedef int  v8i_ __attribute__((ext_vector_type(8)));
typedef int  v4i_ __attribute__((ext_vector_type(4)));

#if __has_builtin(__builtin_amdgcn_tensor_load_to_lds) && __has_builtin(__builtin_amdgcn_s_wait_tensorcnt)
#define HAS_TDM 1
#else
#define HAS_TDM 0
#endif

// ---------------------------------------------------------------------------
// Kernel 1: props_feats gather. One block per (b, s, frame); 125 float4 per row.
// ---------------------------------------------------------------------------
__global__ void __launch_bounds__(128)
tan_gather_kernel(const float* __restrict__ frames,
                  const float* __restrict__ props,
                  const float* __restrict__ duration,
                  float* __restrict__ out) {
    const int blk = blockIdx.x;        // B*S*NF blocks
    const int f   = blk & 15;
    const int bs  = blk >> 4;
    const int b   = bs >> 4;
    const int tid = threadIdx.x;

    const float dur = duration[b];
    const float p0  = props[bs * 2 + 0];
    const float p1  = props[bs * 2 + 1];
    const bool zero = (p0 == 0.0f) && (p1 == 0.0f) && (dur == 0.0f);

    int idx = 0;
    if (dur != 0.0f) {
        int s_fr = min((int)(p0 / dur * 256.0f), T_ - 1);
        int e_fr = min((int)(p1 / dur * 256.0f), T_ - 1);
        // lin = s_fr + f * (e_fr - s_fr) / 16  (float math, matches reference)
        float lin = (float)s_fr + (float)f * (float)(e_fr - s_fr) * (1.0f / 16.0f);
        idx = min((int)lin, T_ - 1);
        idx = max(idx, 0);
    }

    if (tid < 125) {  // 500 floats = 125 float4 (2000B rows are 16B aligned)
        const float4* src = (const float4*)(frames + ((size_t)b * T_ + idx) * D_);
        float4 v;
        if (zero) {
            v = make_float4(0.0f, 0.0f, 0.0f, 0.0f);
        } else {
            v = src[tid];
        }
        ((float4*)(out + ((size_t)bs * NF_ + f) * D_))[tid] = v;
    }
}

// ---------------------------------------------------------------------------
// Kernel 2: IoU overlaps (B,S,N,N). Block-uniform (b,s) -> scalar loads;
// each thread writes one b128 store (4 consecutive j).
// ---------------------------------------------------------------------------
__global__ void __launch_bounds__(256)
tan_overlaps_kernel(const float* __restrict__ y_t,
                    const float* __restrict__ duration,
                    float* __restrict__ out) {
    const int blk   = blockIdx.x;       // B*S*N/4 = 65536 blocks
    const int bs    = blk >> 6;         // uniform per block
    const int iBase = (blk & 63) << 2;  // 4 i-rows per block
    const int tid   = threadIdx.x;
    const int i     = iBase + (tid >> 6);
    const int j0    = (tid & 63) << 2;
    const int b     = bs >> 4;

    const float dur = duration[b];
    const float gs  = y_t[bs * 2 + 0];
    const float ge  = y_t[bs * 2 + 1];
    const bool  gz  = (gs == 0.0f) && (ge == 0.0f);

    // ps = i*dur/N : round-once multiply then exact /256
    const float ps = (float)i * dur * (1.0f / 256.0f);

    float4 r;
    float* rp = &r.x;
#pragma unroll
    for (int k = 0; k < 4; ++k) {
        float pe    = (float)(j0 + k + 1) * dur * (1.0f / 256.0f);
        float inter = fmaxf(0.0f, fminf(pe, ge) - fmaxf(ps, gs));
        float uni   = fmaxf(0.0f, fmaxf(pe, ge) - fminf(ps, gs));
        float ov    = inter / (uni + 1e-6f);
        rp[k] = gz ? 0.0f : ov;
    }
    *(float4*)(out + ((size_t)bs * N_ + i) * N_ + j0) = r;
}

// ---------------------------------------------------------------------------
// Kernel 3: per-(b,s) top-3 + sum over 65536 scores, streamed through LDS by
// the Tensor Data Mover (double-buffered 4096-float tiles).
// ---------------------------------------------------------------------------
__device__ __forceinline__ bool tk_better(float v, int i, float bv, int bi) {
    return (v > bv) || (v == bv && i < bi);   // lax.top_k tie-break: lowest index
}

__device__ __forceinline__ void tk_ins3(float v, int i,
                                        float& v0, float& v1, float& v2,
                                        int& i0, int& i1, int& i2) {
    if (tk_better(v, i, v0, i0)) { v2 = v1; i2 = i1; v1 = v0; i1 = i0; v0 = v; i0 = i; }
    else if (tk_better(v, i, v1, i1)) { v2 = v1; i2 = i1; v1 = v; i1 = i; }
    else if (tk_better(v, i, v2, i2)) { v2 = v; i2 = i; }
}

#if HAS_TDM
// Issue one TDM load: 4096 f32 (tile 512x8, stride 512) from gsrc -> LDS byte addr.
// D# packing per CDNA5 ISA sec 8.3 / 8.4.
__device__ __forceinline__ void tdm_load_f32_tile(const float* gsrc, u32 lds_byte) {
    uint64_t ga = (uint64_t)(uintptr_t)gsrc;
    v4u_ g0;
    g0.x = 1u;                                           // count=1, user descriptor
    g0.y = lds_byte;                                     // lds_addr (bytes)
    g0.z = (u32)ga;                                      // global_addr[31:0]
    g0.w = (u32)((ga >> 32) & 0x01FFFFFFu) | (2u << 30); // global_addr[56:32] | type=2
    v8i_ g1;
    g1[0] = (int)(2u << 16);          // workgroup_mask=0, data_size=2 (4 bytes)
    g1[1] = (int)(512u << 16);        // tensor_dim0 = 512 (low 16 in bits 63:48)
    g1[2] = (int)(8u << 16);          // tensor_dim0 hi=0 | tensor_dim1 = 8
    g1[3] = (int)(512u << 16);        // tensor_dim1 hi=0 | tile_dim0 = 512
    g1[4] = 8;                        // tile_dim1 = 8, tile_dim2 = 0
    g1[5] = 512;                      // tensor_dim0_stride = 512 (low 32)
    g1[6] = 0;                        // stride0 hi | stride1 lo
    g1[7] = 0;                        // stride1 hi
    v4i_ z4 = {0, 0, 0, 0};
    v8i_ z8 = {0, 0, 0, 0, 0, 0, 0, 0};
    // amdgpu-toolchain (clang-23) 6-arg form: (g0, g1, g2, g3, extra, cpol)
    __builtin_amdgcn_tensor_load_to_lds(g0, g1, z4, z4, z8, 0);
}
#endif

__global__ void __launch_bounds__(256)
tan_topk_kernel(const float* __restrict__ scores,
                const float* __restrict__ duration,
                float* __restrict__ max_ts,
                float* __restrict__ max_sc) {
    __shared__ __align__(16) float tile[2][4096];   // 32 KB double buffer (TDM dest)
    __shared__ float rv[256][3];
    __shared__ int   ri[256][3];
    __shared__ float rs[256];

    const int tid = threadIdx.x;
    const int bs  = blockIdx.x;                      // B*S blocks
    const float* base = scores + (size_t)bs * (N_ * N_);

    float v0 = -1e30f, v1 = -1e30f, v2 = -1e30f;
    int   i0 = 0x7fffffff, i1 = 0x7fffffff, i2 = 0x7fffffff;
    float sum = 0.0f;

#if HAS_TDM
    // --- TDM double-buffered streaming: global -> LDS -> VGPR ---
    if (tid == 0) {
        tdm_load_f32_tile(base, (u32)(uintptr_t)&tile[0][0]);
    }
    for (int t = 0; t < 16; ++t) {
        if (tid == 0) {
            if (t + 1 < 16) {
                tdm_load_f32_tile(base + (size_t)(t + 1) * 4096,
                                  (u32)(uintptr_t)&tile[(t + 1) & 1][0]);
                __builtin_amdgcn_s_wait_tensorcnt(1);  // tile t complete (in-order)
            } else {
                __builtin_amdgcn_s_wait_tensorcnt(0);
            }
        }
        __syncthreads();                 // release consumers on completed tile
        const float* buf = tile[t & 1];
#pragma unroll
        for (int k = 0; k < 16; ++k) {   // stride-256: bank-conflict-free
            int   j  = tid + (k << 8);
            float v  = buf[j];
            int   gi = (t << 12) + j;
            sum += v;
            tk_ins3(v, gi, v0, v1, v2, i0, i1, i2);
        }
        __syncthreads();                 // done with buf before it is reloaded
    }
#else
    // --- Fallback: direct coalesced b32 streaming loads ---
    for (int k = 0; k < 256; ++k) {
        int   gi = tid + (k << 8);
        float v  = base[gi];
        sum += v;
        tk_ins3(v, gi, v0, v1, v2, i0, i1, i2);
    }
#endif

    // --- log2 tree merge of per-thread top-3 lists + sum ---
    rv[tid][0] = v0; rv[tid][1] = v1; rv[tid][2] = v2;
    ri[tid][0] = i0; ri[tid][1] = i1; ri[tid][2] = i2;
    rs[tid] = sum;
    __syncthreads();
    for (int st = 128; st > 0; st >>= 1) {
        if (tid < st) {
            rs[tid] += rs[tid + st];
            float a0 = rv[tid][0], a1 = rv[tid][1], a2 = rv[tid][2];
            int   b0 = ri[tid][0], b1 = ri[tid][1], b2 = ri[tid][2];
#pragma unroll
            for (int k = 0; k < 3; ++k) {
                tk_ins3(rv[tid + st][k], ri[tid + st][k], a0, a1, a2, b0, b1, b2);
            }
            rv[tid][0] = a0; rv[tid][1] = a1; rv[tid][2] = a2;
            ri[tid][0] = b0; ri[tid][1] = b1; ri[tid][2] = b2;
        }
        __syncthreads();
    }

    if (tid == 0) {
        const float dur   = duration[bs >> 4];
        const bool  valid = rs[0] > 0.001f;
#pragma unroll
        for (int k = 0; k < TOPK_; ++k) {
            float val = rv[0][k];
            int   idx = ri[0][k];
            float pi  = (float)(idx >> 8);    // idx / N
            float pj  = (float)(idx & 255);   // idx % N
            max_sc[bs * TOPK_ + k]           = valid ? val : 0.0f;
            max_ts[(bs * TOPK_ + k) * 2 + 0] = valid ? pi * (1.0f / 256.0f) * dur : 0.0f;
            max_ts[(bs * TOPK_ + k) * 2 + 1] = valid ? pj * (1.0f / 256.0f) * dur : 0.0f;
        }
    }
}

// ---------------------------------------------------------------------------
// Launch
// ---------------------------------------------------------------------------
extern "C" void kernel_launch(void* const* d_in, const int* in_sizes, int n_in,
                              void* d_out, int out_size, void* d_ws, size_t ws_size,
                              hipStream_t stream) {
    const float* frames   = (const float*)d_in[0];  // (B,T,D)
    const float* props    = (const float*)d_in[1];  // (B,S,2)
    const float* duration = (const float*)d_in[2];  // (B,)
    const float* y_t      = (const float*)d_in[3];  // (B,S,1,2)
    const float* scores   = (const float*)d_in[4];  // (B,S,1,N,N)

    float* out = (float*)d_out;
    // Output layout (concatenated flat, reference return order):
    float* props_feats = out;                             // B*S*NF*D   = 8,192,000
    float* overlaps    = out + (size_t)8192000;           // B*S*N*N    = 67,108,864
    float* max_ts      = out + (size_t)75300864;          // B*S*TOPK*2 = 6,144
    float* max_sc      = out + (size_t)75307008;          // B*S*TOPK   = 3,072

    tan_gather_kernel<<<B_ * S_ * NF_, 128, 0, stream>>>(frames, props, duration, props_feats);
    tan_overlaps_kernel<<<B_ * S_ * N_ / 4, 256, 0, stream>>>(y_t, duration, overlaps);
    tan_topk_kernel<<<B_ * S_, 256, 0, stream>>>(scores, duration, max_ts, max_sc);
}